// Attention_1090921693811
// MI455X (gfx1250) — compile-verified
//
#include <hip/hip_runtime.h>
#include <hip/hip_bf16.h>
#include <math.h>

// ---- Problem constants (fixed by the reference) ----
#define BATCH   8
#define DIM     192
#define HEADS   4
#define CH      48          // DIM / HEADS
#define NPIX    16384       // 128*128
#define QKV_C   576         // 3*DIM
#define IMG     128

typedef float v2f __attribute__((ext_vector_type(2)));
typedef float v8f __attribute__((ext_vector_type(8)));

// D = A(16x4,f32) * B(4x16,f32) + C(16x16,f32), full fp32 matrix pipe.
// A frag (lane L): m=L&15, kb=(L>>4)*2 -> {A[m][kb], A[m][kb+1]}
// B frag (lane L): n=L&15, kb=(L>>4)*2 -> {B[kb][n], B[kb+1][n]}
// C/D (lane L, vgpr r): m = r + (L>=16)*8, n = L&15
__device__ __forceinline__ v8f wmma4(v2f a, v2f b, v8f c) {
  return __builtin_amdgcn_wmma_f32_16x16x4_f32(false, a, false, b, (short)0, c,
                                               false, false);
}

// =====================================================================
// Kernel 1/7: conv1x1 as GEMM  Y[b][m][p] = sum_k W[m][k] * X[b][k][p]
// K fixed at 192. Block = 256 threads = 8 waves; each wave computes a
// 32(M) x 64(N) tile (2x4 grid of 16x16 accums) so every B fragment is
// reused by two WMMAs (8 loads -> 8 WMMAs per K-step). W tile (32x192)
// staged in LDS with stride 193 to avoid bank conflicts.
// =====================================================================
__global__ __launch_bounds__(256) void k_gemm192(
    const float* __restrict__ W, const float* __restrict__ X,
    float* __restrict__ Y, int M) {
  constexpr int K = DIM;
  __shared__ float wT[32 * 193];
  const int b = blockIdx.z;
  const float* Xb = X + (size_t)b * K * NPIX;
  float* Yb = Y + (size_t)b * M * NPIX;
  const int m0 = blockIdx.y * 32;
  const int tid = threadIdx.x;

  for (int i = tid; i < 32 * K; i += 256) {
    int m = i / K, k = i % K;
    wT[m * 193 + k] = W[(size_t)(m0 + m) * K + k];
  }
  __syncthreads();

  const int wave = tid >> 5, lane = tid & 31;
  const int ml = lane & 15, kb = (lane >> 4) << 1;
  const int n0 = blockIdx.x * 512 + wave * 64;

  v8f acc[2][4] = {};
  for (int k0 = 0; k0 < K; k0 += 4) {
    v2f a[2];
#pragma unroll
    for (int t = 0; t < 2; ++t) {
      a[t].x = wT[(t * 16 + ml) * 193 + k0 + kb];
      a[t].y = wT[(t * 16 + ml) * 193 + k0 + kb + 1];
    }
    const float* xr0 = Xb + (size_t)(k0 + kb) * NPIX;
    const float* xr1 = xr0 + NPIX;
    v2f bb[4];
#pragma unroll
    for (int j = 0; j < 4; ++j) {
      int n = n0 + j * 16 + ml;
      bb[j].x = xr0[n];
      bb[j].y = xr1[n];
    }
#pragma unroll
    for (int t = 0; t < 2; ++t)
#pragma unroll
      for (int j = 0; j < 4; ++j)
        acc[t][j] = wmma4(a[t], bb[j], acc[t][j]);
  }
  const int mbase = (lane >> 4) << 3;
#pragma unroll
  for (int t = 0; t < 2; ++t) {
#pragma unroll
    for (int j = 0; j < 4; ++j) {
      int n = n0 + j * 16 + ml;
#pragma unroll
      for (int r = 0; r < 8; ++r)
        Yb[(size_t)(m0 + t * 16 + mbase + r) * NPIX + n] = acc[t][j][r];
    }
  }
}

// =====================================================================
// Kernel 2: depthwise 3x3, SAME zero padding. One block = 256 pixels of
// one (b, channel) plane; 9 weights staged in LDS.
// =====================================================================
__global__ __launch_bounds__(256) void k_dwconv(
    const float* __restrict__ in, const float* __restrict__ wdw,
    float* __restrict__ out) {
  __shared__ float wt[9];
  const int blk = blockIdx.x;
  const int plane = blk >> 6;           // b*576 + c
  const int c = plane % QKV_C;
  const int px = ((blk & 63) << 8) + threadIdx.x;
  if (threadIdx.x < 9) wt[threadIdx.x] = wdw[c * 9 + threadIdx.x];
  __syncthreads();
  const int y = px >> 7, x = px & 127;
  const float* p = in + (size_t)plane * NPIX;
  float s = 0.f;
#pragma unroll
  for (int dy = -1; dy <= 1; ++dy) {
    int yy = y + dy;
    if (yy < 0 || yy > IMG - 1) continue;
#pragma unroll
    for (int dx = -1; dx <= 1; ++dx) {
      int xx = x + dx;
      if (xx < 0 || xx > IMG - 1) continue;
      s += wt[(dy + 1) * 3 + dx + 1] * p[yy * IMG + xx];
    }
  }
  out[(size_t)plane * NPIX + px] = s;
}

// =====================================================================
// Kernel 3: per-row L2 norm scales for q and k. 3072 rows of 16384.
// scales[0..1535] = 1/max(||q_row||,eps); scales[1536..3071] for k.
// =====================================================================
__global__ __launch_bounds__(256) void k_norm(
    const float* __restrict__ qkv, float* __restrict__ scales) {
  __shared__ float red[256];
  const int id = blockIdx.x;              // [0,3072)
  const int qk = id / 1536, rem = id % 1536;
  const int b = rem / DIM, ch = rem % DIM;
  const float* row = qkv + ((size_t)b * QKV_C + qk * DIM + ch) * NPIX;
  float s = 0.f;
  for (int i = threadIdx.x; i < NPIX; i += 256) {
    float v = row[i];
    s += v * v;
  }
  red[threadIdx.x] = s;
  __syncthreads();
  for (int off = 128; off; off >>= 1) {
    if (threadIdx.x < off) red[threadIdx.x] += red[threadIdx.x + off];
    __syncthreads();
  }
  if (threadIdx.x == 0)
    scales[id] = 1.0f / fmaxf(sqrtf(red[0]), 1e-12f);
}

// =====================================================================
// Kernel 4: attn[48][48] = q * k^T over K=16384, per (b,head).
// 8 waves split the K range; each holds a 3x3 grid of 16x16 WMMA accums
// (9 WMMAs per 6 float2 loads); partials merged through LDS ds_add_f32
// atomics, then scaled by normalization factors and temperature.
// =====================================================================
__global__ __launch_bounds__(256) void k_attn(
    const float* __restrict__ qkv, const float* __restrict__ scales,
    const float* __restrict__ temperature, float* __restrict__ attnW) {
  __shared__ float as[CH * CH];
  const int bh = blockIdx.x;
  const int b = bh >> 2, head = bh & 3;
  const int tid = threadIdx.x;
  for (int i = tid; i < CH * CH; i += 256) as[i] = 0.f;
  __syncthreads();

  const float* qb = qkv + ((size_t)b * QKV_C + head * CH) * NPIX;
  const float* kp = qkv + ((size_t)b * QKV_C + DIM + head * CH) * NPIX;
  const int wave = tid >> 5, lane = tid & 31;
  const int ml = lane & 15, kb2 = (lane >> 4) << 1;

  v8f acc[3][3] = {};
  const int kstart = wave * (NPIX / 8);
  for (int kk = kstart; kk < kstart + NPIX / 8; kk += 4) {
    v2f afr[3], bfr[3];
#pragma unroll
    for (int t = 0; t < 3; ++t) {
      const float* qa = qb + (size_t)(t * 16 + ml) * NPIX + kk + kb2;
      afr[t].x = qa[0];
      afr[t].y = qa[1];
      const float* ka = kp + (size_t)(t * 16 + ml) * NPIX + kk + kb2;
      bfr[t].x = ka[0];
      bfr[t].y = ka[1];
    }
#pragma unroll
    for (int mi = 0; mi < 3; ++mi)
#pragma unroll
      for (int di = 0; di < 3; ++di)
        acc[mi][di] = wmma4(afr[mi], bfr[di], acc[mi][di]);
  }
  const int mbase = (lane >> 4) << 3;
#pragma unroll
  for (int mi = 0; mi < 3; ++mi)
#pragma unroll
    for (int di = 0; di < 3; ++di)
#pragma unroll
      for (int r = 0; r < 8; ++r)
        atomicAdd(&as[(mi * 16 + mbase + r) * CH + di * 16 + ml],
                  acc[mi][di][r]);
  __syncthreads();

  const float temp = temperature[head];
  const float* sq = scales + b * DIM + head * CH;
  const float* sk = scales + 1536 + b * DIM + head * CH;
  for (int i = tid; i < CH * CH; i += 256) {
    int m = i / CH, d = i % CH;
    attnW[(size_t)bh * (CH * CH) + i] = as[i] * sq[m] * sk[d] * temp;
  }
}

// =====================================================================
// Kernel 5: 4x top-k masked softmax, combined:
// finalA = sum_i w_i * softmax(mask_topk_i(attn)).
// rank[d] = #{e: a[e]>a[d]} + #{e<d: a[e]==a[d]} matches jax top_k tie-break.
// =====================================================================
__global__ __launch_bounds__(64) void k_smax(
    const float* __restrict__ attnW, const float* __restrict__ a1,
    const float* __restrict__ a2, const float* __restrict__ a3,
    const float* __restrict__ a4, float* __restrict__ finalA) {
  __shared__ float s[CH * CH];
  const int bh = blockIdx.x, tid = threadIdx.x;
  for (int i = tid; i < CH * CH; i += 64)
    s[i] = attnW[(size_t)bh * (CH * CH) + i];
  __syncthreads();
  if (tid >= CH) return;

  const float wgt[4] = {a1[0], a2[0], a3[0], a4[0]};
  const int tk[4] = {CH / 2, CH * 2 / 3, CH * 3 / 4, CH * 4 / 5};  // 24,32,36,38
  const float* row = s + tid * CH;

  int rank[CH];
  for (int d = 0; d < CH; ++d) {
    float ad = row[d];
    int r = 0;
    for (int e = 0; e < CH; ++e) {
      float ae = row[e];
      r += (ae > ad) || (ae == ad && e < d);
    }
    rank[d] = r;
  }
  float out[CH];
  for (int d = 0; d < CH; ++d) out[d] = 0.f;
  for (int v = 0; v < 4; ++v) {
    const int kk = tk[v];
    float m = -3.402823466e38f;
    for (int d = 0; d < CH; ++d)
      if (rank[d] < kk) m = fmaxf(m, row[d]);
    float ssum = 0.f;
    for (int d = 0; d < CH; ++d)
      if (rank[d] < kk) ssum += __expf(row[d] - m);
    const float inv = wgt[v] / ssum;
    for (int d = 0; d < CH; ++d)
      if (rank[d] < kk) out[d] += __expf(row[d] - m) * inv;
  }
  for (int d = 0; d < CH; ++d)
    finalA[(size_t)bh * (CH * CH) + tid * CH + d] = out[d];
}

// =====================================================================
// Kernel 6: outA[48][16384] = finalA[48][48] * v[48][16384] per (b,head).
// finalA staged in LDS (stride 49 vs banks); each wave does a 48x32 tile
// (3x2 accums) so each A fragment feeds two WMMAs.
// =====================================================================
__global__ __launch_bounds__(256) void k_outgemm(
    const float* __restrict__ finalA, const float* __restrict__ qkv,
    float* __restrict__ outA) {
  __shared__ float fa[CH * 49];
  const int bh = blockIdx.y;
  const int b = bh >> 2, head = bh & 3;
  const int tid = threadIdx.x;
  for (int i = tid; i < CH * CH; i += 256) {
    int m = i / CH, k = i % CH;
    fa[m * 49 + k] = finalA[(size_t)bh * (CH * CH) + i];
  }
  __syncthreads();

  const int wave = tid >> 5, lane = tid & 31;
  const int ml = lane & 15, kb2 = (lane >> 4) << 1;
  const int p0 = (blockIdx.x * 8 + wave) * 32;
  const float* vb = qkv + ((size_t)b * QKV_C + 2 * DIM + head * CH) * NPIX;

  v8f acc[3][2] = {};
  for (int k0 = 0; k0 < CH; k0 += 4) {
    v2f bb[2];
#pragma unroll
    for (int j = 0; j < 2; ++j) {
      bb[j].x = vb[(size_t)(k0 + kb2) * NPIX + p0 + j * 16 + ml];
      bb[j].y = vb[(size_t)(k0 + kb2 + 1) * NPIX + p0 + j * 16 + ml];
    }
#pragma unroll
    for (int mi = 0; mi < 3; ++mi) {
      v2f a;
      a.x = fa[(mi * 16 + ml) * 49 + k0 + kb2];
      a.y = fa[(mi * 16 + ml) * 49 + k0 + kb2 + 1];
#pragma unroll
      for (int j = 0; j < 2; ++j)
        acc[mi][j] = wmma4(a, bb[j], acc[mi][j]);
    }
  }
  float* ob = outA + ((size_t)b * DIM + head * CH) * NPIX;
  const int mbase = (lane >> 4) << 3;
#pragma unroll
  for (int mi = 0; mi < 3; ++mi)
#pragma unroll
    for (int j = 0; j < 2; ++j)
#pragma unroll
      for (int r = 0; r < 8; ++r)
        ob[(size_t)(mi * 16 + mbase + r) * NPIX + p0 + j * 16 + ml] =
            acc[mi][j][r];
}

// =====================================================================
// Host-side launch. Workspace layout (bytes):
//   [0)            qkvPre : 8*576*16384*4 = 301,989,888  (reused as outA)
//   [302M)         qkv    : 301,989,888
//   [604M)         scales : 12,288
//   [604M+12K)     attnW  : 294,912
//   [...)          finalA : 294,912     -- total ~604.6 MB
// =====================================================================
extern "C" void kernel_launch(void* const* d_in, const int* in_sizes, int n_in,
                              void* d_out, int out_size, void* d_ws,
                              size_t ws_size, hipStream_t stream) {
  const float* x      = (const float*)d_in[0];
  const float* w_qkv  = (const float*)d_in[1];
  const float* w_dw   = (const float*)d_in[2];
  const float* w_proj = (const float*)d_in[3];
  const float* temp   = (const float*)d_in[4];
  const float* a1     = (const float*)d_in[5];
  const float* a2     = (const float*)d_in[6];
  const float* a3     = (const float*)d_in[7];
  const float* a4     = (const float*)d_in[8];
  float* out = (float*)d_out;

  char* ws = (char*)d_ws;
  const size_t qkvBytes = (size_t)BATCH * QKV_C * NPIX * sizeof(float);
  float* qkvPre = (float*)ws;
  float* qkv    = (float*)(ws + qkvBytes);
  float* scales = (float*)(ws + 2 * qkvBytes);
  float* attnW  = (float*)(ws + 2 * qkvBytes + 12288);
  float* finalA = (float*)(ws + 2 * qkvBytes + 12288 + 294912);
  float* outA   = qkvPre;  // qkvPre dead after dwconv

  // 1) qkv = conv1x1(x): [576x192]*[192x16384] per batch
  k_gemm192<<<dim3(NPIX / 512, QKV_C / 32, BATCH), 256, 0, stream>>>(
      w_qkv, x, qkvPre, QKV_C);
  // 2) depthwise 3x3
  k_dwconv<<<dim3(BATCH * QKV_C * (NPIX / 256)), 256, 0, stream>>>(
      qkvPre, w_dw, qkv);
  // 3) q,k row norms
  k_norm<<<dim3(2 * BATCH * DIM), 256, 0, stream>>>(qkv, scales);
  // 4) attn = (q/||q||)(k/||k||)^T * temperature
  k_attn<<<dim3(BATCH * HEADS), 256, 0, stream>>>(qkv, scales, temp, attnW);
  // 5) combined top-k masked softmax
  k_smax<<<dim3(BATCH * HEADS), 64, 0, stream>>>(attnW, a1, a2, a3, a4,
                                                 finalA);
  // 6) out = finalA * v
  k_outgemm<<<dim3(NPIX / 256, BATCH * HEADS), 256, 0, stream>>>(finalA, qkv,
                                                                 outA);
  // 7) proj conv1x1: [192x192]*[192x16384] per batch
  k_gemm192<<<dim3(NPIX / 512, DIM / 32, BATCH), 256, 0, stream>>>(
      w_proj, outA, out, DIM);
}